// LogSigLayer_87333864997627
// MI455X (gfx1250) — compile-verified
//
#include <hip/hip_runtime.h>
#include <hip/hip_bf16.h>
#include <stddef.h>

// ---------------------------------------------------------------------------
// Log-signature (D=8, M=4, B=128, S=512) for MI455X / gfx1250.
//  - time axis chunked: 16 chunks x 32 steps, per-chunk signatures combined
//    later via Chen associativity (removes the 511-long serial dependence).
//  - level-4 accumulation recast as three K=32 GEMMs per chunk and computed
//    with V_WMMA_F32_16X16X4_F32 (fp32 matrix pipe, full precision).
//  - X1/P staging padded to 16 lanes so WMMA fragment loads are uniform
//    (no exec-mask save/restore around ds_loads); tile loops unrolled so the
//    scheduler can pipeline LDS loads under WMMA issue.
//  - Lyndon basis change done by on-device basis construction + sparse
//    unitriangular forward substitution (A has <=8 nonzeros per row).
// ---------------------------------------------------------------------------

typedef float v2f __attribute__((ext_vector_type(2)));
typedef float v8f __attribute__((ext_vector_type(8)));

__device__ __forceinline__ v8f wmma_f32_k4(v2f a, v2f b, v8f c) {
  // D = A(16x4,f32) * B(4x16,f32) + C(16x16,f32)
  return __builtin_amdgcn_wmma_f32_16x16x4_f32(
      /*neg_a=*/false, a, /*neg_b=*/false, b,
      /*c_mod=*/(short)0, c, /*reuse_a=*/false, /*reuse_b=*/false);
}

// ---------------- geometry constants ----------------
#define BATCH   128
#define STEPS   512
#define NINC    511
#define NCH     16          // chunks per batch
#define TC      32          // time steps per chunk (last one has 31 valid)
#define SIGSZ   4680        // 8 + 64 + 512 + 4096
#define O1      0
#define O2      8
#define O3      72
#define O4      584

// staging slot layout (floats), one slot per time step inside a chunk.
// X1 and P are padded to 16 entries (upper 8 zero) so WMMA fragment loads
// need no lane predication.
#define SX1     0           // dx               (16, [8:15]=0)
#define SP      16          // S1 prefix        (16, [8:15]=0)
#define SX2     32          // dx^2/2           (64)
#define SQ      96          // S2 prefix        (64)
#define SX3     160         // dx^3/6           (512)
#define SR      672         // S3 prefix + dx^3/24 (512)
#define SLOT    1184
#define STATE0  (TC*SLOT)   // S1|S2|S3 running state (584 floats)
#define LDSF    (STATE0 + 584)   // total dynamic LDS floats = 38472 (~150 KB)

// Lyndon basis sizes for D=8, lengths 1..4
#define NROWS   1212        // 8 + 28 + 168 + 1008

// ===========================================================================
// Kernel 1: chunked Chen scan + WMMA level-4 GEMMs
// grid = BATCH*NCH blocks, 256 threads (8 waves)
// ===========================================================================
__global__ void k_scan(const float* __restrict__ path,
                       float* __restrict__ chunkSig) {
  extern __shared__ float sm[];
  const int tid  = threadIdx.x;
  const int lane = tid & 31;
  const int wave = tid >> 5;
  const int blk  = blockIdx.x;
  const int b    = blk >> 4;       // batch
  const int c    = blk & 15;       // chunk

  // zero running state S1|S2|S3
  for (int i = tid; i < 584; i += 256) sm[STATE0 + i] = 0.0f;
  __syncthreads();

  const float* pb = path + (size_t)b * STEPS * 8;
  const int t0 = c * TC;

  for (int k = 0; k < TC; ++k) {
    const int t = t0 + k;
    float* slot = sm + k * SLOT;

    // phase 1: dx -> X1 (padded), copy prefixes P=S1 (padded), Q=S2
    if (tid < 16) {
      float dx = 0.0f;
      if (tid < 8 && t < NINC) dx = pb[(t + 1) * 8 + tid] - pb[t * 8 + tid];
      slot[SX1 + tid] = dx;
      slot[SP + tid]  = (tid < 8) ? sm[STATE0 + tid] : 0.0f;
      if (tid < 8 && t + 4 < STEPS)
        __builtin_prefetch(&pb[(t + 4) * 8 + tid], 0, 1);
    }
    if (tid < 64) slot[SQ + tid] = sm[STATE0 + 8 + tid];
    __syncthreads();

    // phase 2: X2 = dx^2/2, X3 = dx^3/6
    if (tid < 64)
      slot[SX2 + tid] = 0.5f * slot[SX1 + (tid >> 3)] * slot[SX1 + (tid & 7)];
    for (int m = tid; m < 512; m += 256) {
      float v = slot[SX1 + (m >> 6)] * slot[SX1 + ((m >> 3) & 7)] *
                slot[SX1 + (m & 7)];
      slot[SX3 + m] = v * (1.0f / 6.0f);
    }
    __syncthreads();

    // phase 3: R' = S3 + dx^3/24, then update S3,S2,S1 (prefix-safe)
    for (int m = tid; m < 512; m += 256) {
      float s3 = sm[STATE0 + 72 + m];
      float x3 = slot[SX3 + m];
      slot[SR + m] = s3 + 0.25f * x3;
      sm[STATE0 + 72 + m] =
          s3 + slot[SQ + (m >> 3)] * slot[SX1 + (m & 7)] +
          slot[SP + (m >> 6)] * slot[SX2 + (m & 63)] + x3;
    }
    if (tid < 64)
      sm[STATE0 + 8 + tid] +=
          slot[SP + (tid >> 3)] * slot[SX1 + (tid & 7)] + slot[SX2 + tid];
    if (tid < 8) sm[STATE0 + tid] += slot[SX1 + tid];
    __syncthreads();
  }

  // write S1|S2|S3 of this chunk
  float* sig = chunkSig + ((size_t)b * NCH + c) * SIGSZ;
  for (int i = tid; i < 584; i += 256) sig[i] = sm[STATE0 + i];
  float* S4 = sig + O4;

  const int col   = lane & 15;
  const int rbase = (lane < 16) ? 0 : 8;
  const int koff  = (lane < 16) ? 0 : 2;   // A/B k-pair selected by half-wave

  // ---- GEMM (a): S4[m(512), d(8)] = sum_k R'[k][m] * X1[k][d]  (store) ----
#pragma unroll
  for (int ti = 0; ti < 4; ++ti) {
    const int m0 = (wave + 8 * ti) * 16;
    v8f acc; for (int r = 0; r < 8; ++r) acc[r] = 0.0f;
#pragma unroll
    for (int kk = 0; kk < 8; ++kk) {
      const int kb = kk * 4 + koff;
      const int mr = m0 + col;
      v2f a, bf;
      a.x  = sm[(kb + 0) * SLOT + SR + mr];
      a.y  = sm[(kb + 1) * SLOT + SR + mr];
      bf.x = sm[(kb + 0) * SLOT + SX1 + col];   // padded: col in [0,16)
      bf.y = sm[(kb + 1) * SLOT + SX1 + col];
      acc = wmma_f32_k4(a, bf, acc);
    }
    if (col < 8)
      for (int r = 0; r < 8; ++r) S4[(m0 + rbase + r) * 8 + col] = acc[r];
  }
  __threadfence();
  __syncthreads();

  // ---- GEMM (b): S4[ab(64), cd(64)] += sum_k Q[k][ab] * X2[k][cd] ----
#pragma unroll
  for (int ti = 0; ti < 2; ++ti) {
    const int tp = wave + 8 * ti;
    const int m0 = (tp >> 2) * 16, n0 = (tp & 3) * 16;
    v8f acc; for (int r = 0; r < 8; ++r) acc[r] = 0.0f;
#pragma unroll
    for (int kk = 0; kk < 8; ++kk) {
      const int kb = kk * 4 + koff;
      v2f a, bf;
      a.x  = sm[(kb + 0) * SLOT + SQ + m0 + col];
      a.y  = sm[(kb + 1) * SLOT + SQ + m0 + col];
      bf.x = sm[(kb + 0) * SLOT + SX2 + n0 + col];
      bf.y = sm[(kb + 1) * SLOT + SX2 + n0 + col];
      acc = wmma_f32_k4(a, bf, acc);
    }
    for (int r = 0; r < 8; ++r) {
      const int idx = (m0 + rbase + r) * 64 + n0 + col;
      S4[idx] += acc[r];
    }
  }
  __threadfence();
  __syncthreads();

  // ---- GEMM (c): S4[a(8), bcd(512)] += sum_k P[k][a] * X3[k][n] ----
#pragma unroll
  for (int ti = 0; ti < 4; ++ti) {
    const int n0 = (wave + 8 * ti) * 16;
    v8f acc; for (int r = 0; r < 8; ++r) acc[r] = 0.0f;
#pragma unroll
    for (int kk = 0; kk < 8; ++kk) {
      const int kb = kk * 4 + koff;
      v2f a, bf;
      a.x  = sm[(kb + 0) * SLOT + SP + col];    // padded: col in [0,16)
      a.y  = sm[(kb + 1) * SLOT + SP + col];
      bf.x = sm[(kb + 0) * SLOT + SX3 + n0 + col];
      bf.y = sm[(kb + 1) * SLOT + SX3 + n0 + col];
      acc = wmma_f32_k4(a, bf, acc);
    }
    for (int r = 0; r < 8; ++r) {
      const int row = rbase + r;               // a-index; rows >=8 are zero
      if (row < 8) S4[row * 512 + n0 + col] += acc[r];
    }
  }
}

// ===========================================================================
// Kernel 2: fold 16 chunk signatures with Chen, then tensor log (closed form)
// grid = BATCH blocks, 256 threads
// ===========================================================================
__global__ void k_combine_log(const float* __restrict__ chunkSig,
                              float* __restrict__ logsOut) {
  __shared__ float acc[SIGSZ];
  const int tid = threadIdx.x;
  const int b   = blockIdx.x;
  const float* base = chunkSig + (size_t)b * NCH * SIGSZ;

  for (int i = tid; i < SIGSZ; i += 256) acc[i] = base[i];  // chunk 0
  __syncthreads();

  for (int c = 1; c < NCH; ++c) {
    const float* g = base + (size_t)c * SIGSZ;
    float r4[16], r3[2], r2 = 0.0f, r1 = 0.0f;
    for (int q = 0; q < 16; ++q) {
      const int m = tid + 256 * q;
      r4[q] = acc[O4 + m] + g[O4 + m]
            + acc[O1 + (m >> 9)] * g[O3 + (m & 511)]
            + acc[O2 + (m >> 6)] * g[O2 + (m & 63)]
            + acc[O3 + (m >> 3)] * g[O1 + (m & 7)];
    }
    for (int q = 0; q < 2; ++q) {
      const int m = tid + 256 * q;
      r3[q] = acc[O3 + m] + g[O3 + m]
            + acc[O1 + (m >> 6)] * g[O2 + (m & 63)]
            + acc[O2 + (m >> 3)] * g[O1 + (m & 7)];
    }
    if (tid < 64)
      r2 = acc[O2 + tid] + g[O2 + tid] +
           acc[O1 + (tid >> 3)] * g[O1 + (tid & 7)];
    if (tid < 8) r1 = acc[O1 + tid] + g[O1 + tid];
    __syncthreads();
    for (int q = 0; q < 16; ++q) acc[O4 + tid + 256 * q] = r4[q];
    for (int q = 0; q < 2; ++q)  acc[O3 + tid + 256 * q] = r3[q];
    if (tid < 64) acc[O2 + tid] = r2;
    if (tid < 8)  acc[O1 + tid] = r1;
    __syncthreads();
  }

  // tensor log, expanded in compositions (no scratch tensors needed)
  float* L = logsOut + (size_t)b * SIGSZ;
  if (tid < 8) L[O1 + tid] = acc[O1 + tid];
  if (tid < 64)
    L[O2 + tid] = acc[O2 + tid] -
                  0.5f * acc[O1 + (tid >> 3)] * acc[O1 + (tid & 7)];
  for (int q = 0; q < 2; ++q) {
    const int m = tid + 256 * q;
    const float s1a = acc[O1 + (m >> 6)], s1b = acc[O1 + ((m >> 3) & 7)],
                s1c = acc[O1 + (m & 7)];
    L[O3 + m] = acc[O3 + m]
              - 0.5f * (s1a * acc[O2 + (m & 63)] + acc[O2 + (m >> 3)] * s1c)
              + (1.0f / 3.0f) * s1a * s1b * s1c;
  }
  for (int q = 0; q < 16; ++q) {
    const int m = tid + 256 * q;
    const float s0 = acc[O1 + (m >> 9)],       s1 = acc[O1 + ((m >> 6) & 7)];
    const float s2 = acc[O1 + ((m >> 3) & 7)], s3 = acc[O1 + (m & 7)];
    float t = acc[O4 + m]
        - 0.5f * (s0 * acc[O3 + (m & 511)] +
                  acc[O2 + (m >> 6)] * acc[O2 + (m & 63)] +
                  acc[O3 + (m >> 3)] * s3)
        + (1.0f / 3.0f) * (s0 * s1 * acc[O2 + (m & 63)] +
                           s0 * acc[O2 + ((m >> 3) & 63)] * s3 +
                           acc[O2 + (m >> 6)] * s2 * s3)
        - 0.25f * s0 * s1 * s2 * s3;
    L[O4 + m] = t;
  }
}

// ===========================================================================
// Lyndon basis construction (device side)
// ===========================================================================
__device__ __forceinline__ int lexless(const int* a, int la,
                                       const int* b, int lb) {
  const int n = (la < lb) ? la : lb;
  for (int i = 0; i < n; ++i) {
    if (a[i] < b[i]) return 1;
    if (a[i] > b[i]) return 0;
  }
  return la < lb;
}

// sorted Lyndon codes per level, packed: [0,8) len1 | [8,36) len2 |
// [36,204) len3 | [204,1212) len4
__global__ void k_lyndon_lists(int* __restrict__ lists) {
  if (threadIdx.x != 0 || blockIdx.x != 0) return;
  int pos = 0;
  for (int len = 1; len <= 4; ++len) {
    int total = 1;
    for (int i = 0; i < len; ++i) total *= 8;
    for (int code = 0; code < total; ++code) {
      int w[4];
      for (int i = 0; i < len; ++i) w[i] = (code >> (3 * (len - 1 - i))) & 7;
      bool ly = true;
      for (int s = 1; s < len && ly; ++s)
        if (!lexless(w, len, w + s, len - s)) ly = false;
      if (ly) lists[pos++] = code;
    }
  }
}

struct EList { int n; int code[8]; float co[8]; };
__constant__ int P8c[5] = {1, 8, 64, 512, 4096};

__device__ void eadd(EList& e, int c, float v) {
  for (int i = 0; i < e.n; ++i)
    if (e.code[i] == c) { e.co[i] += v; return; }
  e.code[e.n] = c; e.co[e.n] = v; e.n++;
}
__device__ void ecomb(EList& o, const EList& eu, int lu,
                      const EList& ev, int lv) {
  o.n = 0;
  for (int i = 0; i < eu.n; ++i)
    for (int j = 0; j < ev.n; ++j) {
      const float p = eu.co[i] * ev.co[j];
      eadd(o, eu.code[i] * P8c[lv] + ev.code[j],  p);
      eadd(o, ev.code[j] * P8c[lu] + eu.code[i], -p);
    }
}
__device__ void expand1(int c, EList& o) { o.n = 1; o.code[0] = c; o.co[0] = 1.0f; }
__device__ void expand2(const int* w, EList& o) {
  EList a, b; expand1(w[0], a); expand1(w[1], b); ecomb(o, a, 1, b, 1);
}
__device__ int findSplit(const int* w, int len) {  // start of min proper suffix
  int best = 1;
  for (int s = 2; s < len; ++s)
    if (lexless(w + s, len - s, w + best, len - best)) best = s;
  return best;
}
__device__ void expand_small(const int* w, int len, EList& o) {
  if (len == 1) expand1(w[0], o); else expand2(w, o);
}
__device__ void expand3(const int* w, EList& o) {
  const int p = findSplit(w, 3);
  EList a, b;
  expand_small(w, p, a);
  expand_small(w + p, 3 - p, b);
  ecomb(o, a, p, b, 3 - p);
}
__device__ void expand4(const int* w, EList& o) {
  const int p = findSplit(w, 4);
  EList a, b;
  if (p <= 2) expand_small(w, p, a); else expand3(w, a);
  const int lv = 4 - p;
  if (lv <= 2) expand_small(w + p, lv, b); else expand3(w + p, b);
  ecomb(o, a, p, b, lv);
}

// per-row sparse A: diag coeff + off-diagonal (j, coeff), j level-local
__global__ void k_rows(const int* __restrict__ lists,
                       int* __restrict__ nterm, int* __restrict__ termJ,
                       float* __restrict__ termC, float* __restrict__ diagC) {
  const int r = blockIdx.x * blockDim.x + threadIdx.x;
  if (r >= NROWS) return;
  int lvl, off, n;
  if (r < 8)        { lvl = 1; off = 0;   n = 8;    }
  else if (r < 36)  { lvl = 2; off = 8;   n = 28;   }
  else if (r < 204) { lvl = 3; off = 36;  n = 168;  }
  else              { lvl = 4; off = 204; n = 1008; }
  const int j0 = r - off;
  const int code = lists[off + j0];
  int w[4];
  for (int i = 0; i < lvl; ++i) w[i] = (code >> (3 * (lvl - 1 - i))) & 7;

  EList e;
  if (lvl == 1) expand1(w[0], e);
  else if (lvl == 2) expand2(w, e);
  else if (lvl == 3) expand3(w, e);
  else expand4(w, e);

  int cnt = 0;
  float dg = 1.0f;
  for (int t = 0; t < e.n; ++t) {
    int lo = 0, hi = n - 1, found = -1;
    while (lo <= hi) {
      const int mid = (lo + hi) >> 1;
      const int v = lists[off + mid];
      if (v == e.code[t]) { found = mid; break; }
      if (v < e.code[t]) lo = mid + 1; else hi = mid - 1;
    }
    if (found < 0) continue;             // non-Lyndon word: ignored
    if (found == j0) { dg = e.co[t]; continue; }
    termJ[r * 8 + cnt] = found;
    termC[r * 8 + cnt] = e.co[t];
    ++cnt;
  }
  nterm[r] = cnt;
  diagC[r] = dg;
}

// ===========================================================================
// Kernel 5: gather Lyndon coords + sparse unitriangular substitution
// out . A = lyn  =>  forward scatter per level. 1 block, 128 threads (=batch)
// ===========================================================================
__global__ void k_project(const float* __restrict__ logs,
                          const int* __restrict__ lists,
                          const int* __restrict__ nterm,
                          const int* __restrict__ termJ,
                          const float* __restrict__ termC,
                          const float* __restrict__ diagC,
                          float* __restrict__ out) {
  const int b = threadIdx.x;
  if (b >= BATCH) return;
  const float* L = logs + (size_t)b * SIGSZ;
  float* o = out + (size_t)b * NROWS;
  const int nl[4]   = {8, 28, 168, 1008};
  const int ooff[4] = {0, 8, 36, 204};
  const int loff[4] = {O1, O2, O3, O4};
  for (int k = 0; k < 4; ++k) {
    const int n = nl[k], oo = ooff[k], lo = loff[k];
    for (int j = 0; j < n; ++j) o[oo + j] = L[lo + lists[oo + j]];
    for (int i = 0; i < n; ++i) {
      const int r = oo + i;
      const float xi = o[oo + i] / diagC[r];
      o[oo + i] = xi;
      const int c = nterm[r];
      for (int t = 0; t < c; ++t)
        o[oo + termJ[r * 8 + t]] -= xi * termC[r * 8 + t];
    }
  }
}

// ===========================================================================
extern "C" void kernel_launch(void* const* d_in, const int* in_sizes, int n_in,
                              void* d_out, int out_size, void* d_ws,
                              size_t ws_size, hipStream_t stream) {
  (void)in_sizes; (void)n_in; (void)out_size; (void)ws_size;
  const float* path = (const float*)d_in[0];
  float* out = (float*)d_out;

  // workspace carve-up (all 4-byte units)
  float* wsf = (float*)d_ws;
  const size_t CH = (size_t)BATCH * NCH * SIGSZ;   // chunk signatures
  const size_t LG = (size_t)BATCH * SIGSZ;         // log-signatures
  float* chunkSig = wsf;
  float* logsBuf  = wsf + CH;
  int*   lists    = (int*)(wsf + CH + LG);         // 1212 ints
  int*   nterm    = lists + NROWS;                 // 1212
  int*   termJ    = nterm + NROWS;                 // 1212*8
  float* termC    = (float*)(termJ + NROWS * 8);   // 1212*8
  float* diagC    = termC + NROWS * 8;             // 1212

  k_lyndon_lists<<<1, 64, 0, stream>>>(lists);
  k_rows<<<(NROWS + 127) / 128, 128, 0, stream>>>(lists, nterm, termJ, termC,
                                                  diagC);
  const size_t smBytes = (size_t)LDSF * sizeof(float);  // ~150 KB dynamic LDS
  k_scan<<<BATCH * NCH, 256, smBytes, stream>>>(path, chunkSig);
  k_combine_log<<<BATCH, 256, 0, stream>>>(chunkSig, logsBuf);
  k_project<<<1, 128, 0, stream>>>(logsBuf, lists, nterm, termJ, termC, diagC,
                                   out);
}